// MetaR_867583394073
// MI455X (gfx1250) — compile-verified
//
#include <hip/hip_runtime.h>

typedef float v2f __attribute__((ext_vector_type(2)));
typedef float v8f __attribute__((ext_vector_type(8)));

constexpr int NNEI = 200;   // neighbors per batch row
constexpr int DIM  = 100;   // embedding dim
constexpr int K3   = 300;   // 3*DIM contracted dim
constexpr int LDH  = 104;   // padded LDS row stride (floats)
constexpr int MT   = 13;    // ceil(200/16) row tiles
constexpr int NT   = 7;     // ceil(100/16) col tiles
constexpr int NTILES = MT * NT;  // 91

// LDS layout (float offsets)
constexpr int OFF_H    = 0;
constexpr int OFF_LOG  = OFF_H + NNEI * LDH;   // 20800
constexpr int OFF_OUTA = OFF_LOG + 208;
constexpr int OFF_AW   = OFF_OUTA + 104;
constexpr int OFF_GW   = OFF_AW + 104;
constexpr int OFF_BIAS = OFF_GW + 104;
constexpr int OFF_SCR  = OFF_BIAS + 104;
constexpr int SMEM_FLOATS = OFF_SCR + 8;       // ~85.8 KB — fits CDNA5's 320KB/WGP LDS

__global__ __launch_bounds__(256)
void gcn_attn_kernel(const int* __restrict__ conn,      // (2048, 200, 4)
                     const float* __restrict__ emb,     // (100001, 100)
                     const float* __restrict__ gcn_w,   // (100, 300)
                     const float* __restrict__ gcn_w_bias,
                     const float* __restrict__ gcn_b,
                     const float* __restrict__ attn_w,  // (100,)
                     const float* __restrict__ attn_bias,
                     const float* __restrict__ gate_w,  // (100,)
                     const float* __restrict__ gate_w_bias,
                     const float* __restrict__ gate_b,
                     float* __restrict__ out)           // (2048, 100)
{
    extern __shared__ float smem[];
    float* Hs   = smem + OFF_H;
    float* logi = smem + OFF_LOG;
    float* outa = smem + OFF_OUTA;
    float* aws  = smem + OFF_AW;
    float* gws  = smem + OFF_GW;
    float* bias = smem + OFF_BIAS;
    float* scr  = smem + OFF_SCR;

    const int b    = blockIdx.x;
    const int tid  = threadIdx.x;
    const int wid  = tid >> 5;
    const int lane = tid & 31;
    const int l    = lane & 15;   // row-within-tile / col-within-tile
    const int half = lane >> 4;   // K half selector for f32 WMMA fragments

    // Preload small vectors into LDS
    if (tid < DIM) {
        bias[tid] = gcn_w_bias[tid] + gcn_b[tid];
        aws[tid]  = attn_w[tid];
        gws[tid]  = gate_w[tid];
    }
    __syncthreads();

    // -------- Stage 1: H = leaky_relu(concat @ gcn_w^T + bias) via v_wmma_f32_16x16x4_f32
    for (int tile = wid; tile < NTILES; tile += 8) {
        const int mt  = tile / NT;
        const int nt  = tile - mt * NT;
        const int row = mt * 16 + l;
        const int rr  = row < NNEI ? row : (NNEI - 1);     // clamp pad rows (ignored later)
        const int4 ci = *(const int4*)(conn + ((size_t)b * NNEI + rr) * 4);
        const float* p0 = emb + (size_t)ci.y * DIM;        // relation embedding row
        const float* p1 = emb + (size_t)ci.z * DIM;        // entity embedding row
        const float* p2 = emb + (size_t)ci.w * DIM;        // time embedding row
        const int dcol = nt * 16 + l;
        const int dc   = dcol < DIM ? dcol : (DIM - 1);    // clamp pad cols (ignored later)
        const float* wrow = gcn_w + (size_t)dc * K3;

        v8f c = {0.f, 0.f, 0.f, 0.f, 0.f, 0.f, 0.f, 0.f};
        const float* ap[3] = {p0, p1, p2};
        #pragma unroll
        for (int s = 0; s < 3; ++s) {
            // A frag: lane holds A[M=l][k0 + 2*half + {0,1}]  (f32 16x4 layout)
            // B frag: lane holds B[k0 + 2*half + {0,1}][N=l] = gcn_w[dc][k]
            const float* ab = ap[s] + half * 2;
            const float* bb = wrow + s * DIM + half * 2;
            #pragma unroll 5
            for (int kk = 0; kk < 25; ++kk) {
                v2f a  = *(const v2f*)(ab + kk * 4);       // 8B gather (L2-resident table)
                v2f bv = *(const v2f*)(bb + kk * 4);       // 8B cached weight load
                c = __builtin_amdgcn_wmma_f32_16x16x4_f32(
                        false, a, false, bv, (short)0, c, false, false);
            }
        }

        // C/D layout: lane l (+16*half), VGPR i -> element (m = half*8 + i, n = l)
        const int colg = nt * 16 + l;
        if (colg < DIM) {
            const float bsum  = bias[colg];
            const int   mbase = mt * 16 + half * 8;
            #pragma unroll
            for (int i = 0; i < 8; ++i) {
                const int m = mbase + i;
                if (m < NNEI) {
                    float v = c[i] + bsum;
                    Hs[m * LDH + colg] = (v > 0.f) ? v : 0.01f * v;   // leaky_relu(0.01)
                }
            }
        }
    }
    __syncthreads();

    // -------- Stage 2a: attention logits (one thread per neighbor)
    if (tid < NNEI) {
        const float* hrow = Hs + tid * LDH;
        float acc = 0.f;
        #pragma unroll 4
        for (int d = 0; d < DIM; ++d) acc += hrow[d] * aws[d];
        logi[tid] = acc + attn_bias[0];
    }
    __syncthreads();

    // -------- Stage 2b: softmax over 200 neighbors (wave 0)
    if (tid < 32) {
        float mx = -3.4e38f;
        for (int n = tid; n < NNEI; n += 32) mx = fmaxf(mx, logi[n]);
        #pragma unroll
        for (int o = 16; o > 0; o >>= 1) mx = fmaxf(mx, __shfl_xor(mx, o, 32));
        float sm = 0.f;
        for (int n = tid; n < NNEI; n += 32) {
            float e = __expf(logi[n] - mx);
            logi[n] = e;
            sm += e;
        }
        #pragma unroll
        for (int o = 16; o > 0; o >>= 1) sm += __shfl_xor(sm, o, 32);
        if (tid == 0) scr[0] = 1.f / sm;
    }
    __syncthreads();

    // -------- Stage 2c: out_attn[d] = sum_n w[n] * H[n][d] (one thread per dim)
    if (tid < DIM) {
        float acc = 0.f;
        #pragma unroll 4
        for (int n = 0; n < NNEI; ++n) acc += logi[n] * Hs[n * LDH + tid];
        outa[tid] = acc * scr[0];
    }
    __syncthreads();

    // -------- Stage 2d: gate scalar (wave 0)
    if (tid < 32) {
        float g = 0.f;
        for (int d = tid; d < DIM; d += 32) g += outa[d] * gws[d];
        #pragma unroll
        for (int o = 16; o > 0; o >>= 1) g += __shfl_xor(g, o, 32);
        if (tid == 0) {
            g += gate_w_bias[0] + gate_b[0];
            scr[1] = 1.f / (1.f + __expf(-g));
        }
    }
    __syncthreads();

    // -------- Stage 2e: blend with self embedding
    if (tid < DIM) {
        const float g    = scr[1];
        const int   self = conn[(size_t)b * NNEI * 4];   // connections[b,0,0]
        const float se   = emb[(size_t)self * DIM + tid];
        out[(size_t)b * DIM + tid] = outa[tid] * g + se * (1.f - g);
    }
}

extern "C" void kernel_launch(void* const* d_in, const int* in_sizes, int n_in,
                              void* d_out, int out_size, void* d_ws, size_t ws_size,
                              hipStream_t stream) {
    const int*   conn = (const int*)d_in[0];
    // d_in[1] = num_neighbors: unused by the reference computation (no masking)
    const float* emb  = (const float*)d_in[2];
    const float* gw   = (const float*)d_in[3];
    const float* gwb  = (const float*)d_in[4];
    const float* gb   = (const float*)d_in[5];
    const float* aw   = (const float*)d_in[6];
    const float* ab   = (const float*)d_in[7];
    const float* gtw  = (const float*)d_in[8];
    const float* gtwb = (const float*)d_in[9];
    const float* gtb  = (const float*)d_in[10];
    float* out = (float*)d_out;

    const size_t smem_bytes = (size_t)SMEM_FLOATS * sizeof(float);  // ~85.8 KB
    hipLaunchKernelGGL(gcn_attn_kernel, dim3(2048), dim3(256), smem_bytes, stream,
                       conn, emb, gw, gwb, gb, aw, ab, gtw, gtwb, gtb, out);
}